// AltBatchSplitFF_59880434041152
// MI455X (gfx1250) — compile-verified
//
#include <hip/hip_runtime.h>

// ---------------------------------------------------------------------------
// AltBatchSplitFF on MI455X (gfx1250): dispatch-first MoE-style grouped FF.
// All matmuls run on V_WMMA_F32_16X16X4_F32 (exact fp32 tensor op).
//
// Shapes: B=4 S=2048 D=512 E=16 SETS=4 -> ES=64 slots, F=32, T=16 tokens/group,
// NG = B*S/T = 512 groups. x/out: [8192,512] f32 row-major.
// controller: [512,64]  (d, es)      f1: [512,64,32] (d, es, f)
// f2: [64,32,512] (es, f, d)  -- each slot's f2 is a contiguous 64KB chunk.
// ---------------------------------------------------------------------------

typedef float v2f __attribute__((ext_vector_type(2)));
typedef float v8f __attribute__((ext_vector_type(8)));

#define D_DIM 512
#define ES    64
#define F_DIM 32
#define T_TOK 16
#define NG    512            // total groups = B*S/T
#define APAD  516            // 16-row A-tile row stride (floats): 516%64=4 -> conflict-free
#define IPAD  34             // inner-tile row stride: even (8B align) + distinct banks

static __device__ __forceinline__ v8f wmma4(v2f a, v2f b, v8f c) {
  // D = A(16x4 f32) * B(4x16 f32) + C(16x16 f32)
  return __builtin_amdgcn_wmma_f32_16x16x4_f32(
      /*neg_a=*/false, a, /*neg_b=*/false, b,
      /*c_mod=*/(short)0, c, /*reuse_a=*/false, /*reuse_b=*/false);
}

// ---------------------------------------------------------------------------
// Kernel 1: router. One block (128 thr = 4 waves) per group.
// logits[16 x 64] = Xg[16 x 512] @ controller[512 x 64], then per-slot argmax
// over the 16 tokens with the reference's linspace tie-breaker.
// ---------------------------------------------------------------------------
__global__ void __launch_bounds__(128)
router_kernel(const float* __restrict__ x, const float* __restrict__ ctrl,
              int* __restrict__ ridx) {
  __shared__ __align__(16) float Xg[T_TOK * APAD];
  __shared__ float lg[T_TOK * ES];

  const int group = blockIdx.x;            // 0..511
  const int tid   = threadIdx.x;           // 0..127

  // Stage 16x512 token tile (coalesced float4), rows padded to 516 floats.
  const float* xbase = x + (size_t)group * T_TOK * D_DIM;
  for (int p = tid; p < T_TOK * (D_DIM / 4); p += 128) {
    const int row = p >> 7;                // /128 float4 per row
    const int c4  = p & 127;
    ((float4*)&Xg[row * APAD])[c4] = ((const float4*)(xbase + row * D_DIM))[c4];
  }
  __syncthreads();

  const int wave = tid >> 5, lane = tid & 31;
  const int hh = lane >> 4, ln = lane & 15;
  const int es0 = wave * 16;               // each wave owns a 16-slot N-tile

  v8f c = {};
  for (int k0 = 0; k0 < D_DIM; k0 += 4) {
    const int k = k0 + 2 * hh;
    v2f a; a.x = Xg[ln * APAD + k];        a.y = Xg[ln * APAD + k + 1];
    v2f b; b.x = ctrl[(size_t)k * ES + es0 + ln];
           b.y = ctrl[(size_t)(k + 1) * ES + es0 + ln];
    c = wmma4(a, b, c);
  }
  for (int r = 0; r < 8; ++r) {            // C layout: VGPR r -> rows r, r+8
    const int M = r + 8 * hh;
    lg[M * ES + es0 + ln] = c[r];
  }
  __syncthreads();

  if (tid < ES) {                          // one thread per slot
    float best = -3.4e38f; int bi = 0;
    for (int t = 0; t < T_TOK; ++t) {
      const float v = lg[t * ES + tid] + (float)t * (1e-6f / 15.0f);
      if (v > best) { best = v; bi = t; }
    }
    ridx[group * ES + tid] = group * T_TOK + bi;   // flat routed token row
  }
}

// ---------------------------------------------------------------------------
// Kernel 2: zero the output (unrouted tokens stay 0; harness poisons d_out).
// ---------------------------------------------------------------------------
__global__ void zero_kernel(float4* __restrict__ out, int n4) {
  for (int i = blockIdx.x * blockDim.x + threadIdx.x; i < n4;
       i += gridDim.x * blockDim.x)
    out[i] = make_float4(0.f, 0.f, 0.f, 0.f);
}

// ---------------------------------------------------------------------------
// Kernel 3: expert path, slot-major so both projections are dense GEMMs.
// grid = (32 M-tiles of 16 groups, 64 slots), block = 128 thr = 4 waves.
//   A-tile[16x512] (gathered routed rows) @ f1_es[512x32] -> relu -> inner
//   inner[16x32] @ f2_es[32x512] -> atomic scatter-add into out rows.
// ---------------------------------------------------------------------------
__global__ void __launch_bounds__(128)
expert_kernel(const float* __restrict__ x, const float* __restrict__ f1,
              const float* __restrict__ f2, const int* __restrict__ ridx,
              float* __restrict__ out) {
  __shared__ __align__(16) float At[T_TOK * APAD];     // 33.0 KB
  __shared__ __align__(16) float W1[D_DIM * F_DIM];    // 64  KB
  __shared__ float In[T_TOK * IPAD];                   // 2.1 KB
  __shared__ int   rIdx[T_TOK];

  const int mt  = blockIdx.x;              // 0..31  (M tile: 16 groups)
  const int es  = blockIdx.y;              // 0..63  (expert slot)
  const int tid = threadIdx.x;

  if (tid < T_TOK) rIdx[tid] = ridx[(mt * T_TOK + tid) * ES + es];
  __syncthreads();

  // Gather 16 routed rows into the A-tile (8 threads per row, coalesced).
  {
    const int row = tid >> 3, sub = tid & 7;
    const float* src = x + (size_t)rIdx[row] * D_DIM;
    for (int q = 0; q < 16; ++q) {
      const int c4 = sub + 8 * q;
      ((float4*)&At[row * APAD])[c4] = ((const float4*)src)[c4];
    }
  }
  // Stage this slot's f1 (512x32, row stride 64*32 in global) into LDS.
  {
    const float* w1g = f1 + (size_t)es * F_DIM;
    for (int p = tid; p < D_DIM * (F_DIM / 4); p += 128) {
      const int d = p >> 3, f4 = p & 7;
      ((float4*)W1)[p] = *(const float4*)(w1g + (size_t)d * (ES * F_DIM) + f4 * 4);
    }
  }
  __syncthreads();

  const int wave = tid >> 5, lane = tid & 31;
  const int hh = lane >> 4, ln = lane & 15;
  const float* w2g = f2 + (size_t)es * (F_DIM * D_DIM); // contiguous 64KB slot

  if (wave < 2) {
    // GEMM2: 16x512 @ 512x32. Wave w owns F-tile [16w, 16w+16).
    const int f0 = wave * 16;
    v8f c = {};
    for (int k0 = 0; k0 < D_DIM; k0 += 4) {
      const int k = k0 + 2 * hh;
      v2f a; a.x = At[ln * APAD + k];      a.y = At[ln * APAD + k + 1];
      v2f b; b.x = W1[k * F_DIM + f0 + ln]; b.y = W1[(k + 1) * F_DIM + f0 + ln];
      c = wmma4(a, b, c);
    }
    for (int r = 0; r < 8; ++r) {          // ReLU into inner tile
      const int M = r + 8 * hh;
      const float v = c[r];
      In[M * IPAD + f0 + ln] = v > 0.f ? v : 0.f;
    }
  } else {
    // Waves 2-3: warm f2_es into cache while GEMM2 runs (global_prefetch_b8).
    const char* p2 = (const char*)w2g;
    for (int t = tid - 64; t < 512; t += 64)
      __builtin_prefetch(p2 + (size_t)t * 128, 0, 1);
  }
  __syncthreads();

  // GEMM3: 16x32 @ 32x512. Each wave covers 8 of the 32 D-tiles.
  for (int q = 0; q < 8; ++q) {
    const int d0 = (wave + 4 * q) * 16;
    v8f c = {};
    for (int k0 = 0; k0 < F_DIM; k0 += 4) {
      const int k = k0 + 2 * hh;
      v2f a; a.x = In[ln * IPAD + k];      a.y = In[ln * IPAD + k + 1];
      v2f b; b.x = w2g[(size_t)k * D_DIM + d0 + ln];
             b.y = w2g[(size_t)(k + 1) * D_DIM + d0 + ln];
      c = wmma4(a, b, c);
    }
    for (int r = 0; r < 8; ++r) {          // combine: scatter-add to routed row
      const int M  = r + 8 * hh;
      const int ro = rIdx[M];
      atomicAdd(&out[(size_t)ro * D_DIM + d0 + ln], c[r]);
    }
  }
}

// ---------------------------------------------------------------------------
extern "C" void kernel_launch(void* const* d_in, const int* in_sizes, int n_in,
                              void* d_out, int out_size, void* d_ws, size_t ws_size,
                              hipStream_t stream) {
  const float* x    = (const float*)d_in[0];   // [4,2048,512]
  const float* ctrl = (const float*)d_in[1];   // [512,16,4]  == [512,64]
  const float* f1   = (const float*)d_in[2];   // [512,16,4,32]
  const float* f2   = (const float*)d_in[3];   // [16,4,32,512]
  float* out = (float*)d_out;
  int*   ridx = (int*)d_ws;                    // 512*64 ints = 128 KB scratch

  zero_kernel<<<2048, 256, 0, stream>>>((float4*)out, (NG * T_TOK * D_DIM) / 4);
  router_kernel<<<NG, 128, 0, stream>>>(x, ctrl, ridx);
  expert_kernel<<<dim3(NG / T_TOK, ES), 128, 0, stream>>>(x, f1, f2, ridx, out);
}